// AttentionLayer_44873818309057
// MI455X (gfx1250) — compile-verified
//
#include <hip/hip_runtime.h>

#define Bn 4
#define Sn 4096
#define Dn 512
#define Hn 512

typedef __attribute__((ext_vector_type(2))) float v2f;
typedef __attribute__((ext_vector_type(4))) float v4f;
typedef __attribute__((ext_vector_type(8))) float v8f;

// ---------------------------------------------------------------------------
// q = query @ Wq + bq   (per batch, H=512 outputs)
// ---------------------------------------------------------------------------
__global__ __launch_bounds__(256) void qproj_kernel(
    const float* __restrict__ query, const float* __restrict__ Wq,
    const float* __restrict__ bq, float* __restrict__ q)
{
    __shared__ float qs[Dn];
    int b = blockIdx.x, t = threadIdx.x;
    qs[t]       = query[b * Dn + t];
    qs[t + 256] = query[b * Dn + t + 256];
    __syncthreads();
    float a0 = bq[t], a1 = bq[t + 256];
    for (int d = 0; d < Dn; ++d) {
        float qd = qs[d];
        a0 += qd * Wq[d * Hn + t];
        a1 += qd * Wq[d * Hn + t + 256];
    }
    q[b * Hn + t]       = a0;
    q[b * Hn + t + 256] = a1;
}

// ---------------------------------------------------------------------------
// t[b,d] = sum_h Wk[d,h]*q[b,h] ;  beta[b] = bk . q[b]
// (so scores = keys . t + beta, avoiding materializing k entirely)
// ---------------------------------------------------------------------------
__global__ __launch_bounds__(256) void tproj_kernel(
    const float* __restrict__ Wk, const float* __restrict__ bk,
    const float* __restrict__ q, float* __restrict__ tv, float* __restrict__ beta)
{
    __shared__ float qs[Hn];
    __shared__ float red[256];
    int b = blockIdx.x, t = threadIdx.x;
    qs[t]       = q[b * Hn + t];
    qs[t + 256] = q[b * Hn + t + 256];
    __syncthreads();
    for (int pass = 0; pass < 2; ++pass) {
        int d = t + pass * 256;
        const float* wrow = Wk + (size_t)d * Hn;
        float acc = 0.f;
        for (int h = 0; h < Hn; ++h) acc += wrow[h] * qs[h];
        tv[b * Dn + d] = acc;
    }
    float pb = bk[t] * qs[t] + bk[t + 256] * qs[t + 256];
    red[t] = pb; __syncthreads();
    for (int o = 128; o > 0; o >>= 1) { if (t < o) red[t] += red[t + o]; __syncthreads(); }
    if (t == 0) beta[b] = red[0];
}

// ---------------------------------------------------------------------------
// scores[b,s] = keys[b,s,:] . t[b,:] + beta[b]   (wave-per-row GEMV)
// ---------------------------------------------------------------------------
__global__ __launch_bounds__(256) void scores_kernel(
    const float* __restrict__ keys, const float* __restrict__ tv,
    const float* __restrict__ beta, float* __restrict__ scores)
{
    int t = threadIdx.x;
    int wave = t >> 5, lane = t & 31;
    int row = blockIdx.x * 8 + wave;          // 0..B*S-1
    int b = row >> 12;                        // S = 4096
    const float* kr = keys + (size_t)row * Dn;
    const float* tb = tv + b * Dn;
    float acc = 0.f;
#pragma unroll
    for (int i = 0; i < Dn / 32; ++i) acc += kr[lane + 32 * i] * tb[lane + 32 * i];
#pragma unroll
    for (int m = 16; m >= 1; m >>= 1) acc += __shfl_xor(acc, m, 32);
    if (lane == 0) scores[row] = acc + beta[b];
}

// ---------------------------------------------------------------------------
// Per batch: gmax, e[s]=exp((score-gmax)/sqrt(H)), inclusive prefix denom,
// invd[r]=1/denom[r].   One block per batch, block-level scan of 4096.
// ---------------------------------------------------------------------------
__global__ __launch_bounds__(256) void softmax_scan_kernel(
    const float* __restrict__ scores, float* __restrict__ e, float* __restrict__ invd)
{
    __shared__ float sc[Sn];
    __shared__ float red[256];
    int b = blockIdx.x, t = threadIdx.x;
    const float* sb = scores + b * Sn;
    for (int i = t; i < Sn; i += 256) sc[i] = sb[i];
    __syncthreads();
    float m = -3.0e38f;
    for (int i = t; i < Sn; i += 256) m = fmaxf(m, sc[i]);
    red[t] = m; __syncthreads();
    for (int o = 128; o > 0; o >>= 1) { if (t < o) red[t] = fmaxf(red[t], red[t + o]); __syncthreads(); }
    float gmax = red[0];
    __syncthreads();
    const float kc = 1.4426950408889634f / 22.627416997969522f;  // log2(e)/sqrt(512)
    int base = t * 16;
    float run = 0.f;
    for (int j = 0; j < 16; ++j) {
        float evv = exp2f((sc[base + j] - gmax) * kc);
        e[b * Sn + base + j] = evv;
        run += evv;
        sc[base + j] = run;                    // inclusive prefix within segment
    }
    red[t] = run; __syncthreads();
    if (t == 0) { float r2 = 0.f; for (int k2 = 0; k2 < 256; ++k2) { float x = red[k2]; red[k2] = r2; r2 += x; } }
    __syncthreads();
    float off = red[t];
    for (int j = 0; j < 16; ++j)
        invd[b * Sn + base + j] = 1.0f / (off + sc[base + j]);
}

// ---------------------------------------------------------------------------
// C[M,N] = A[M,K] @ W[K,N] + bias[N]  via V_WMMA_F32_16X16X4_F32.
// Block: 256 thr / 8 waves, tile 64x64, KC=32 staged in LDS.
// Wave (wm 0..3, wn 0..1) owns a 16x32 sub-tile (two 16x16 accumulators).
// A operand: lane(m=lane&15) holds A[m][2*hi+j]; B operand: lane(n) holds
// B[2*hi+j][n]  ->  B staged transposed Bt[n][k] so both are v2f LDS loads.
// ---------------------------------------------------------------------------
__global__ __launch_bounds__(256) void gemm_wmma_f32(
    const float* __restrict__ A, const float* __restrict__ W,
    const float* __restrict__ bias, float* __restrict__ C,
    int M, int N, int K)
{
    __shared__ float As[64][36];   // row stride 144B: 16B-aligned, conflict-free reads
    __shared__ float Bt[64][36];   // Bt[n][k]
    int t = threadIdx.x;
    int wave = t >> 5, lane = t & 31;
    int wm = wave & 3, wn = wave >> 2;
    int hi = lane >> 4, lm = lane & 15;
    int bm = blockIdx.x * 64, bn = blockIdx.y * 64;

    v8f acc0 = {0.f, 0.f, 0.f, 0.f, 0.f, 0.f, 0.f, 0.f};
    v8f acc1 = acc0;

    int ar = t >> 3;               // 0..31 (A stage: 2 passes of 32 rows)
    int ac = (t & 7) * 4;          // 0..28
    int wr = t >> 4;               // 0..15 (W stage: 2 passes of 16 rows)
    int wc = (t & 15) * 4;         // 0..60

    for (int k0 = 0; k0 < K; k0 += 32) {
        v4f a0 = *(const v4f*)(A + (size_t)(bm + ar) * K + k0 + ac);
        v4f a1 = *(const v4f*)(A + (size_t)(bm + ar + 32) * K + k0 + ac);
        v4f w0 = *(const v4f*)(W + (size_t)(k0 + wr) * N + bn + wc);
        v4f w1 = *(const v4f*)(W + (size_t)(k0 + wr + 16) * N + bn + wc);
        *(v4f*)&As[ar][ac]      = a0;
        *(v4f*)&As[ar + 32][ac] = a1;
        Bt[wc + 0][wr] = w0.x; Bt[wc + 1][wr] = w0.y;
        Bt[wc + 2][wr] = w0.z; Bt[wc + 3][wr] = w0.w;
        Bt[wc + 0][wr + 16] = w1.x; Bt[wc + 1][wr + 16] = w1.y;
        Bt[wc + 2][wr + 16] = w1.z; Bt[wc + 3][wr + 16] = w1.w;
        __syncthreads();

        const float* arow  = &As[wm * 16 + lm][0];
        const float* brow0 = &Bt[wn * 32 + lm][0];
        const float* brow1 = &Bt[wn * 32 + 16 + lm][0];
#pragma unroll
        for (int kk = 0; kk < 32; kk += 4) {
            v2f av  = *(const v2f*)(arow  + kk + 2 * hi);
            v2f bv0 = *(const v2f*)(brow0 + kk + 2 * hi);
            v2f bv1 = *(const v2f*)(brow1 + kk + 2 * hi);
            acc0 = __builtin_amdgcn_wmma_f32_16x16x4_f32(false, av, false, bv0,
                                                         (short)0, acc0, false, false);
            acc1 = __builtin_amdgcn_wmma_f32_16x16x4_f32(false, av, false, bv1,
                                                         (short)0, acc1, false, false);
        }
        __syncthreads();
    }
    int n0 = bn + wn * 32 + lm;
    float b0v = bias[n0], b1v = bias[n0 + 16];
    int r0 = bm + wm * 16 + hi * 8;
#pragma unroll
    for (int i = 0; i < 8; ++i) {
        C[(size_t)(r0 + i) * N + n0]      = acc0[i] + b0v;
        C[(size_t)(r0 + i) * N + n0 + 16] = acc1[i] + b1v;
    }
}

// ---------------------------------------------------------------------------
// Weighted prefix scan over S:  context[b,r,h] = invd[b,r] * sum_{s<=r} e[b,s]*v[b,s,h]
// 3 phases: per-chunk partials -> chunk scan -> apply (in-place over v buffer).
// ---------------------------------------------------------------------------
__global__ __launch_bounds__(128) void ctx_partial_kernel(
    const float* __restrict__ e, const float* __restrict__ v, float* __restrict__ cs)
{
    int chunk = blockIdx.x, b = blockIdx.z;
    int h = blockIdx.y * 128 + threadIdx.x;
    int s0 = chunk * 64;
    const float* eb = e + b * Sn + s0;
    const float* vb = v + ((size_t)(b * Sn + s0)) * Hn + h;
    float acc = 0.f;
    for (int i = 0; i < 64; ++i) acc += eb[i] * vb[(size_t)i * Hn];
    cs[(b * 64 + chunk) * Hn + h] = acc;
}

__global__ __launch_bounds__(256) void ctx_scan_kernel(float* __restrict__ cs)
{
    int b = blockIdx.y;
    int h = blockIdx.x * 256 + threadIdx.x;
    float run = 0.f;
    for (int c = 0; c < 64; ++c) {
        int idx = (b * 64 + c) * Hn + h;
        float x = cs[idx]; cs[idx] = run; run += x;   // exclusive prefix
    }
}

__global__ __launch_bounds__(128) void ctx_apply_kernel(
    const float* __restrict__ e, const float* __restrict__ invd,
    const float* __restrict__ cs, float* __restrict__ v)
{
    int chunk = blockIdx.x, b = blockIdx.z;
    int h = blockIdx.y * 128 + threadIdx.x;
    int s0 = chunk * 64;
    float run = cs[(b * 64 + chunk) * Hn + h];
    const float* eb = e + b * Sn;
    const float* ib = invd + b * Sn;
    float* vb = v + ((size_t)b * Sn) * Hn + h;
    for (int i = 0; i < 64; ++i) {
        int s = s0 + i;
        run += eb[s] * vb[(size_t)s * Hn];
        vb[(size_t)s * Hn] = run * ib[s];             // in-place: v -> context
    }
}

// ---------------------------------------------------------------------------
// attn[b,r,s] = (s<=r) ? e[b,s]*invd[b,r] : 0   (256 MB, nontemporal stores)
// ---------------------------------------------------------------------------
__global__ __launch_bounds__(256) void attn_write_kernel(
    const float* __restrict__ e, const float* __restrict__ invd, float* __restrict__ attn)
{
    int r = blockIdx.x, b = blockIdx.y, t = threadIdx.x;
    float ir = invd[b * Sn + r];
    const float* eb = e + b * Sn;
    float* row = attn + ((size_t)b * Sn + r) * Sn;
#pragma unroll
    for (int pass = 0; pass < 4; ++pass) {
        int s = pass * 1024 + t * 4;
        v4f val;
        val.x = (s + 0 <= r) ? eb[s + 0] * ir : 0.f;
        val.y = (s + 1 <= r) ? eb[s + 1] * ir : 0.f;
        val.z = (s + 2 <= r) ? eb[s + 2] * ir : 0.f;
        val.w = (s + 3 <= r) ? eb[s + 3] * ir : 0.f;
        __builtin_nontemporal_store(val, (v4f*)(row + s));
    }
}

// ---------------------------------------------------------------------------
extern "C" void kernel_launch(void* const* d_in, const int* in_sizes, int n_in,
                              void* d_out, int out_size, void* d_ws, size_t ws_size,
                              hipStream_t stream) {
    const float* query  = (const float*)d_in[0];
    const float* keys   = (const float*)d_in[1];
    const float* values = (const float*)d_in[2];
    const float* Wq = (const float*)d_in[3];
    const float* bq = (const float*)d_in[4];
    const float* Wk = (const float*)d_in[5];
    const float* bk = (const float*)d_in[6];
    const float* Wv = (const float*)d_in[7];
    const float* bv = (const float*)d_in[8];
    const float* Wo = (const float*)d_in[9];
    const float* bo = (const float*)d_in[10];

    float* out  = (float*)d_out;
    float* attn = out + (size_t)Bn * Sn * Dn;      // outputs concatenated: out, attn_weights
    float* ws   = (float*)d_ws;

    float* qv     = ws + 0;        // 2048
    float* tv     = ws + 2048;     // 2048
    float* beta   = ws + 4096;     // 4
    float* scores = ws + 8192;     // 16384
    float* ev     = ws + 24576;    // 16384
    float* invd   = ws + 40960;    // 16384
    float* cs     = ws + 57344;    // 4*64*512 = 131072
    float* vbuf   = ws + 196608;   // 16384*512 = 8388608  (v, then context in-place)

    qproj_kernel<<<dim3(Bn), dim3(256), 0, stream>>>(query, Wq, bq, qv);
    tproj_kernel<<<dim3(Bn), dim3(256), 0, stream>>>(Wk, bk, qv, tv, beta);
    scores_kernel<<<dim3((Bn * Sn) / 8), dim3(256), 0, stream>>>(keys, tv, beta, scores);
    softmax_scan_kernel<<<dim3(Bn), dim3(256), 0, stream>>>(scores, ev, invd);
    gemm_wmma_f32<<<dim3((Bn * Sn) / 64, Hn / 64), dim3(256), 0, stream>>>(
        values, Wv, bv, vbuf, Bn * Sn, Hn, Dn);
    ctx_partial_kernel<<<dim3(64, Hn / 128, Bn), dim3(128), 0, stream>>>(ev, vbuf, cs);
    ctx_scan_kernel<<<dim3(Hn / 256, Bn), dim3(256), 0, stream>>>(cs);
    ctx_apply_kernel<<<dim3(64, Hn / 128, Bn), dim3(128), 0, stream>>>(ev, invd, cs, vbuf);
    gemm_wmma_f32<<<dim3((Bn * Sn) / 64, Dn / 64), dim3(256), 0, stream>>>(
        vbuf, Wo, bo, out, Bn * Sn, Dn, Hn);
    attn_write_kernel<<<dim3(Sn, Bn), dim3(256), 0, stream>>>(ev, invd, attn);
}